// PFMModel_24575802867939
// MI455X (gfx1250) — compile-verified
//
#include <hip/hip_runtime.h>
#include <cstdint>

// ---------------------------------------------------------------------------
// PFM rollout: B=16, A=1000, H=8, N=1024, PRED_LEN=12
// One wave32 per agent; agent's 1024 neighbors (8KB) staged once via the
// CDNA5 async global->LDS data mover (global_load_async_to_lds_b128 +
// s_wait_asynccnt), then cached in registers for all 12 sequential
// timesteps. Both reference scans are fused into one loop.
// VALU-bound => inner loop uses raw v_sqrt_f32 / v_rcp_f32 (1-ulp) and
// split accumulators for dual-issue ILP.
// ---------------------------------------------------------------------------

constexpr int B_ = 16;
constexpr int A_ = 1000;
constexpr int H_ = 8;
constexpr int N_ = 1024;
constexpr int PRED_LEN_ = 12;
constexpr int WAVES_PER_BLOCK = 8;
constexpr int F4_PER_AGENT = N_ / 2;            // 512 float4 (2 neighbors each)
constexpr int F4_PER_LANE  = F4_PER_AGENT / 32; // 16

__global__ __launch_bounds__(WAVES_PER_BLOCK * 32)
void pfm_kernel(const float*  __restrict__ history,    // (B,A,H,2)
                const float4* __restrict__ neighbors4, // (B,A,N/2) as float4
                const float*  __restrict__ goal,       // (2,)
                const float*  __restrict__ coeff,      // (A,3)
                float*        __restrict__ out)        // (B,A,12,2)
{
    __shared__ float4 smem[WAVES_PER_BLOCK * F4_PER_AGENT]; // 64 KB

    const int tid   = threadIdx.x;
    const int wave  = tid >> 5;
    const int lane  = tid & 31;
    const int agent = blockIdx.x * WAVES_PER_BLOCK + wave;  // grid sized exactly
    const int a     = agent % A_;

    // ---- stage this wave's 8KB neighbor tile into LDS with async DMA ------
    float4* sw = smem + wave * F4_PER_AGENT;
    const float4* gbase = neighbors4 + (size_t)agent * F4_PER_AGENT;
    {
        // low 32 bits of a generic pointer to LDS == LDS byte offset
        const uint32_t lds_base = (uint32_t)(uintptr_t)sw;
#pragma unroll
        for (int i = 0; i < F4_PER_LANE; ++i) {
            const uint64_t gaddr = (uint64_t)(uintptr_t)(gbase + i * 32 + lane);
            const uint32_t laddr = lds_base + (uint32_t)((i * 32 + lane) * sizeof(float4));
            asm volatile("global_load_async_to_lds_b128 %0, %1, off"
                         :: "v"(laddr), "v"(gaddr)
                         : "memory");
        }
    }

    // ---- overlap: scalar/per-agent parameter loads ------------------------
    const float k1 = coeff[a * 3 + 0];
    const float k2 = coeff[a * 3 + 1];
    const float kr = coeff[a * 3 + 2];
    const float gx = goal[0];
    const float gy = goal[1];

    // history[-2] and history[-1] are 4 consecutive floats, 16B aligned
    const float4 h2 = *(const float4*)(history + ((size_t)agent * H_ + (H_ - 2)) * 2);
    const float cx0 = h2.z, cy0 = h2.w;           // history[:, :, -1]
    const float velx = h2.z - h2.x, vely = h2.w - h2.y;

    // ---- wait for async DMA, pull tile into registers ---------------------
#if defined(__has_builtin) && __has_builtin(__builtin_amdgcn_s_wait_asynccnt)
    __builtin_amdgcn_s_wait_asynccnt(0);
#else
    asm volatile("s_wait_asynccnt 0" ::: "memory");
#endif

    float4 nb[F4_PER_LANE];
#pragma unroll
    for (int i = 0; i < F4_PER_LANE; ++i)
        nb[i] = sw[i * 32 + lane];                // ds_load_b128

    const float MINS = (float)(4.087 * (1.0 - 0.15)); // 3.47395
    const float MAXS = (float)(4.087 * (1.0 + 0.15)); // 4.70005

    float curx = cx0, cury = cy0;   // scan-1 carry
    float sx = cx0, sy = cy0;       // scan-2 (speed) carry

    for (int t = 0; t < PRED_LEN_; ++t) {
        // --- repulsion over this lane's 32 cached neighbors ---
        // split accumulators: two independent FMA chains for dual-issue ILP
        float fx0 = 0.0f, fy0 = 0.0f;
        float fx1 = 0.0f, fy1 = 0.0f;
#pragma unroll
        for (int i = 0; i < F4_PER_LANE; ++i) {
            const float4 q = nb[i];
            {
                const float dx = curx - q.x, dy = cury - q.y;
                const float d2 = fmaf(dx, dx, dy * dy);
                const float d  = __builtin_amdgcn_sqrtf(d2) + 1e-6f;   // v_sqrt_f32
                const float r  = __builtin_amdgcn_rcpf(d * d);         // v_rcp_f32
                const float w  = (d < 0.5f) ? r : 0.0f;
                fx0 = fmaf(dx, w, fx0);
                fy0 = fmaf(dy, w, fy0);
            }
            {
                const float dx = curx - q.z, dy = cury - q.w;
                const float d2 = fmaf(dx, dx, dy * dy);
                const float d  = __builtin_amdgcn_sqrtf(d2) + 1e-6f;
                const float r  = __builtin_amdgcn_rcpf(d * d);
                const float w  = (d < 0.5f) ? r : 0.0f;
                fx1 = fmaf(dx, w, fx1);
                fy1 = fmaf(dy, w, fy1);
            }
        }
        float fx = fx0 + fx1;
        float fy = fy0 + fy1;

        // --- wave32 butterfly reduction ---
#pragma unroll
        for (int off = 16; off > 0; off >>= 1) {
            fx += __shfl_xor(fx, off, 32);
            fy += __shfl_xor(fy, off, 32);
        }

        // --- forces + integration (scan 1) ---
        float Fx = k1 * (gx - curx) + kr * fx;
        float Fy = k1 * (gy - cury) + kr * fy;
        if (t == 0) { Fx += k2 * velx; Fy += k2 * vely; }
        curx += Fx * 0.1f;
        cury += Fy * 0.1f;

        // --- fused speed-clip step (scan 2): precise math, once per step ---
        const float dx = curx - sx, dy = cury - sy;
        const float sp = sqrtf(dx * dx + dy * dy);
        const float cl = fminf(fmaxf(sp, MINS), MAXS);
        const float spf = (sp > 0.0f) ? cl : sp;
        const float scale = spf / (sp + 1e-8f);
        sx = sx + dx * scale;
        sy = sy + dy * scale;

        if (lane == 0) {
            float2 o; o.x = sx; o.y = sy;
            *(float2*)(out + ((size_t)agent * PRED_LEN_ + t) * 2) = o;
        }
    }
}

extern "C" void kernel_launch(void* const* d_in, const int* in_sizes, int n_in,
                              void* d_out, int out_size, void* d_ws, size_t ws_size,
                              hipStream_t stream) {
    const float*  history   = (const float*)d_in[0];
    const float4* neighbors = (const float4*)d_in[1];
    const float*  goal      = (const float*)d_in[2];
    const float*  coeff     = (const float*)d_in[3];
    float*        out       = (float*)d_out;

    const dim3 grid((B_ * A_) / WAVES_PER_BLOCK); // 2000
    const dim3 block(WAVES_PER_BLOCK * 32);       // 256 threads = 8 wave32
    hipLaunchKernelGGL(pfm_kernel, grid, block, 0, stream,
                       history, neighbors, goal, coeff, out);
}